// ACE_reprsenttaion_56495999812196
// MI455X (gfx1250) — compile-verified
//
#include <hip/hip_runtime.h>
#include <hip/hip_bf16.h>
#include <math.h>

typedef float v2f __attribute__((ext_vector_type(2)));
typedef float v8f __attribute__((ext_vector_type(8)));

// ---------------------------------------------------------------------------
// WMMA wrapper: D = A(16x4 f32) * B(4x16 f32) + C(16x16 f32)
// A layout: lane m=lane&15; lanes 0-15 hold K={0,1}, lanes 16-31 hold K={2,3}
// B layout: lane n=lane&15; lanes 0-15 hold K={0,1}, lanes 16-31 hold K={2,3}
// C/D layout: lane n=lane&15; VGPR j holds M=j (lanes 0-15) / M=j+8 (16-31)
// ---------------------------------------------------------------------------
static __device__ __forceinline__ v8f wmma_f32_16x16x4(v2f a, v2f b, v8f c) {
  return __builtin_amdgcn_wmma_f32_16x16x4_f32(false, a, false, b, (short)0, c,
                                               false, false);
}

// fast silu: x * v_rcp_f32(1 + exp(-x))  (avoids IEEE div expansion)
static __device__ __forceinline__ float fast_silu(float x) {
  return x * __builtin_amdgcn_rcpf(1.0f + __expf(-x));
}

static __device__ __forceinline__ v2f ld2(const float* p) {
  return *(const v2f*)p;  // 8B-aligned LDS pair -> ds_load_b64
}

// ---------------------------------------------------------------------------
// Zero the A accumulator region of the workspace
// ---------------------------------------------------------------------------
__global__ void ace_zero_kernel(float* __restrict__ p, long long n) {
  long long i = (long long)blockIdx.x * blockDim.x + threadIdx.x;
  const long long stride = (long long)gridDim.x * blockDim.x;
  for (; i < n; i += stride) p[i] = 0.0f;
}

// ---------------------------------------------------------------------------
// Real Clebsch-Gordan tables computed on-device (graph-capture safe).
// Layout in ws (floats): CG0 l=0..3 at {0,1,10,35} (sizes 1,9,25,49),
// CG1 pairs (0,1),(1,2),(2,3) at {84,93,138} (sizes 9,45,105). Total 243.
// ---------------------------------------------------------------------------
struct cd { double re, im; };
static __device__ __forceinline__ cd cmul(cd a, cd b) {
  cd r; r.re = a.re * b.re - a.im * b.im; r.im = a.re * b.im + a.im * b.re;
  return r;
}
static __device__ double factd(int n) {
  double r = 1.0;
  for (int i = 2; i <= n; ++i) r *= (double)i;
  return r;
}
static __device__ int imax3(int a, int b, int c) { int m = a > b ? a : b; return m > c ? m : c; }
static __device__ int imin3(int a, int b, int c) { int m = a < b ? a : b; return m < c ? m : c; }

static __device__ double su2cg(int j1, int m1, int j2, int m2, int j3, int m3) {
  if (m3 != m1 + m2) return 0.0;
  const int vmin = imax3(-j1 + j2 + m3, -j1 + m1, 0);
  const int vmax = imin3(j2 + j3 + m1, j3 - j1 + j2, j3 + m3);
  if (vmax < vmin) return 0.0;
  double C = ((double)(2 * j3 + 1) * factd(j3 + j1 - j2) * factd(j3 - j1 + j2) *
              factd(j1 + j2 - j3) * factd(j3 + m3) * factd(j3 - m3)) /
             (factd(j1 + j2 + j3 + 1) * factd(j1 - m1) * factd(j1 + m1) *
              factd(j2 - m2) * factd(j2 + m2));
  C = sqrt(C);
  double S = 0.0;
  for (int v = vmin; v <= vmax; ++v) {
    const double sg = ((v + j2 + m2) & 1) ? -1.0 : 1.0;
    S += sg * factd(j2 + j3 + m1 - v) * factd(j1 - m1 + v) /
         (factd(v) * factd(j3 - j1 + j2 - v) * factd(j3 + m3 - v) *
          factd(v + j1 - j2 - m3));
  }
  return C * S;
}

static __device__ void qmat(int l, cd* q) {
  const int D = 2 * l + 1;
  for (int i = 0; i < D * D; ++i) { q[i].re = 0.0; q[i].im = 0.0; }
  const double is2 = 0.70710678118654752440;
  for (int m = -l; m < 0; ++m) {
    q[(l + m) * D + (l - m)].re = is2;            // col l+|m|
    q[(l + m) * D + (l + m)].im = -is2;           // col l-|m|
  }
  q[l * D + l].re = 1.0;
  for (int m = 1; m <= l; ++m) {
    const double sg = (m & 1) ? -1.0 : 1.0;
    q[(l + m) * D + (l + m)].re = sg * is2;
    q[(l + m) * D + (l - m)].im = sg * is2;
  }
  cd ph;
  switch (l & 3) {                                 // (-i)^l
    case 0:  ph.re = 1.0;  ph.im = 0.0;  break;
    case 1:  ph.re = 0.0;  ph.im = -1.0; break;
    case 2:  ph.re = -1.0; ph.im = 0.0;  break;
    default: ph.re = 0.0;  ph.im = 1.0;  break;
  }
  for (int i = 0; i < D * D; ++i) q[i] = cmul(q[i], ph);
}

__global__ void ace_cg_init_kernel(float* __restrict__ cg) {
  const int t = threadIdx.x;
  if (t >= 7) return;
  int l1, l2, l3, off;
  switch (t) {
    case 0: l1 = 0; l2 = 0; l3 = 0; off = 0;   break;
    case 1: l1 = 1; l2 = 1; l3 = 0; off = 1;   break;
    case 2: l1 = 2; l2 = 2; l3 = 0; off = 10;  break;
    case 3: l1 = 3; l2 = 3; l3 = 0; off = 35;  break;
    case 4: l1 = 0; l2 = 1; l3 = 1; off = 84;  break;
    case 5: l1 = 1; l2 = 2; l3 = 1; off = 93;  break;
    default:l1 = 2; l2 = 3; l3 = 1; off = 138; break;
  }
  cd q1[49], q2[49], q3[49];
  qmat(l1, q1); qmat(l2, q2); qmat(l3, q3);
  const int D1 = 2 * l1 + 1, D2 = 2 * l2 + 1, D3 = 2 * l3 + 1;
  float* out = cg + off;
  // Cr[j,k,n] = sum_{i,kk,m} q1[i,j] q2[kk,k] conj(q3[n,m]) * su2cg(i-l1,kk-l2,m-l3)
  for (int j = 0; j < D1; ++j)
    for (int k = 0; k < D2; ++k)
      for (int n = 0; n < D3; ++n) {
        double sre = 0.0;
        for (int i = 0; i < D1; ++i)
          for (int kk = 0; kk < D2; ++kk)
            for (int m = 0; m < D3; ++m) {
              const double c = su2cg(l1, i - l1, l2, kk - l2, l3, m - l3);
              if (c != 0.0) {
                cd tt = cmul(q1[i * D1 + j], q2[kk * D2 + k]);
                cd q3c; q3c.re = q3[n * D3 + m].re; q3c.im = -q3[n * D3 + m].im;
                tt = cmul(tt, q3c);
                sre += tt.re * c;
              }
            }
        out[(j * D2 + k) * D3 + n] = (float)sre;
      }
}

// ---------------------------------------------------------------------------
// Edge kernel: 8 waves/block, 16 edges/wave. Per-l MLP layers via f32 WMMA,
// then coalesced atomic scatter of R_l * Y_lm * Z_sender into A[N,16,32].
//
// Weights are stored in LDS K-pair-swizzled: element W[k][n] lives at
//   [(k>>1)*64 + n*2 + (k&1)]
// so each lane's B-fragment pair (W[k][n], W[k+1][n]) is one ds_load_b64.
// ---------------------------------------------------------------------------
// LDS float layout:
//   s_w0 [4*256]    @ 0     (swizzled, K=8)
//   s_w1 [4*1024]   @ 1024  (swizzled, K=32)
//   s_w2 [4*1024]   @ 5120  (swizzled, K=32)
//   s_b0 [4*32]     @ 9216
//   s_b1 [4*32]     @ 9344
//   s_b2 [4*32]     @ 9472
//   per-wave (stride 1408) @ 9600: rb[16*8], ylm[16*16], z[16*32], h[16*32]
//   int area @ 20864 floats: per-wave 32 ints (recv[16], zidx[16])
#define EDGE_SMEM_FLOATS 21120
#define EDGE_SMEM_BYTES  (EDGE_SMEM_FLOATS * 4)

static __device__ __forceinline__ void mlp_layer_k32(
    const float* __restrict__ Wswz, const float* __restrict__ Bv,
    float* h_w, int lane, int l15, int klo, int koff, bool do_silu) {
  v8f acc0, acc1;
  {
    const float bb0 = Bv[l15], bb1 = Bv[16 + l15];
#pragma unroll
    for (int j = 0; j < 8; ++j) { acc0[j] = bb0; acc1[j] = bb1; }
  }
#pragma unroll
  for (int s = 0; s < 8; ++s) {
    const float* wrow = Wswz + (s * 2 + koff) * 64;
    const v2f a  = ld2(h_w + l15 * 32 + s * 4 + klo);
    const v2f bA = ld2(wrow + l15 * 2);
    const v2f bB = ld2(wrow + 32 + l15 * 2);
    acc0 = wmma_f32_16x16x4(a, bA, acc0);
    acc1 = wmma_f32_16x16x4(a, bB, acc1);
  }
  const int mo = (lane < 16) ? 0 : 8;
#pragma unroll
  for (int j = 0; j < 8; ++j) {
    float x0 = acc0[j], x1 = acc1[j];
    if (do_silu) {
      x0 = fast_silu(x0);
      x1 = fast_silu(x1);
    }
    h_w[(j + mo) * 32 + l15]      = x0;
    h_w[(j + mo) * 32 + 16 + l15] = x1;
  }
}

__global__ __launch_bounds__(256) void ace_edge_kernel(
    const float* __restrict__ pos, const int* __restrict__ atomic_num,
    const int* __restrict__ edge_index, const float* __restrict__ shifts,
    const float* __restrict__ w_z2k,
    const float* __restrict__ w0, const float* __restrict__ b0,
    const float* __restrict__ w1, const float* __restrict__ b1,
    const float* __restrict__ w2, const float* __restrict__ b2,
    float* __restrict__ A_ws, float* __restrict__ rb_out,
    float* __restrict__ ylm_out, int E) {
  extern __shared__ float smem[];
  float* s_w0 = smem;
  float* s_w1 = smem + 1024;
  float* s_w2 = smem + 5120;
  float* s_b0 = smem + 9216;
  float* s_b1 = smem + 9344;
  float* s_b2 = smem + 9472;

  const int tid = threadIdx.x;
  // K-pair swizzled weight staging: W[k][n] -> [(k>>1)*64 + n*2 + (k&1)]
  for (int i = tid; i < 4 * 8 * 32; i += 256) {
    const int l = i >> 8, r = i & 255, kk = r >> 5, n = r & 31;
    s_w0[l * 256 + (kk >> 1) * 64 + n * 2 + (kk & 1)] = w0[i];
  }
  for (int i = tid; i < 4 * 32 * 32; i += 256) {
    const int l = i >> 10, r = i & 1023, kk = r >> 5, n = r & 31;
    const int d = l * 1024 + (kk >> 1) * 64 + n * 2 + (kk & 1);
    s_w1[d] = w1[i];
    s_w2[d] = w2[i];
  }
  for (int i = tid; i < 4 * 32; i += 256) {
    s_b0[i] = b0[i]; s_b1[i] = b1[i]; s_b2[i] = b2[i];
  }

  const int wave = tid >> 5;
  const int lane = tid & 31;
  const int l15  = lane & 15;
  const int tileBase = blockIdx.x * 128 + wave * 16;

  float* rb_w  = smem + 9600 + wave * 1408;
  float* ylm_w = rb_w + 128;
  float* z_w   = ylm_w + 256;
  float* h_w   = z_w + 512;
  int*   recv_w = (int*)(smem + 20864) + wave * 32;
  int*   zidx_w = recv_w + 16;

  // ---- scalar phase: lanes 0..15, one edge each ----
  if (lane < 16) {
    const int e = tileBase + lane;
    const bool ok = (e < E);
    const int snd = ok ? edge_index[e] : 0;
    const int rcv = ok ? edge_index[E + e] : 0;
    recv_w[lane] = rcv;
    zidx_w[lane] = ok ? atomic_num[snd] : 0;
    const float vx = pos[3 * rcv + 0] - pos[3 * snd + 0] + (ok ? shifts[3 * e + 0] : 0.0f);
    const float vy = pos[3 * rcv + 1] - pos[3 * snd + 1] + (ok ? shifts[3 * e + 1] : 0.0f);
    const float vz = pos[3 * rcv + 2] - pos[3 * snd + 2] + (ok ? shifts[3 * e + 2] : 0.0f);
    float len = sqrtf(vx * vx + vy * vy + vz * vz);
    len = fmaxf(len, 1e-6f);
    const float inv = 1.0f / len;
    const float x = vx * inv, y = vy * inv, z = vz * inv;
    // polynomial cutoff p=6: 1 - 28 x^6 + 48 x^7 - 21 x^8 for x<1
    const float xr = len * 0.2f;
    float cut = 0.0f;
    if (xr < 1.0f) {
      const float x2 = xr * xr, x3 = x2 * xr;
      const float x6 = x3 * x3, x7 = x6 * xr, x8 = x7 * xr;
      cut = 1.0f - 28.0f * x6 + 48.0f * x7 - 21.0f * x8;
    }
    const float rbn = 0.63245553203368f;  // sqrt(2/5)
#pragma unroll
    for (int n = 0; n < 8; ++n) {
      const float v = ok ? (rbn * __sinf((float)(n + 1) * 0.628318530717959f * len) * inv * cut)
                         : 0.0f;
      rb_w[lane * 8 + n] = v;
      if (ok) rb_out[(size_t)e * 8 + n] = v;
    }
    // real spherical harmonics l=0..3 (component normalized)
    float Y[16];
    const float s3 = 1.732050807568877f, s15 = 3.872983346207417f;
    Y[0] = 1.0f;
    Y[1] = s3 * y; Y[2] = s3 * z; Y[3] = s3 * x;
    Y[4] = s15 * x * y; Y[5] = s15 * y * z;
    Y[6] = 1.118033988749895f * (3.0f * z * z - 1.0f);
    Y[7] = s15 * x * z;
    Y[8] = 1.936491673103709f * (x * x - y * y);
    Y[9]  = 2.091650066335189f * y * (3.0f * x * x - y * y);
    Y[10] = 10.24695076595960f * x * y * z;
    Y[11] = 1.620185174601965f * y * (5.0f * z * z - 1.0f);
    Y[12] = 1.322875655532295f * z * (5.0f * z * z - 3.0f);
    Y[13] = 1.620185174601965f * x * (5.0f * z * z - 1.0f);
    Y[14] = 5.123475382979799f * z * (x * x - y * y);
    Y[15] = 2.091650066335189f * x * (x * x - 3.0f * y * y);
#pragma unroll
    for (int i = 0; i < 16; ++i) {
      const float v = ok ? Y[i] : 0.0f;
      ylm_w[lane * 16 + i] = v;
      if (ok) ylm_out[(size_t)e * 16 + i] = v;
    }
  }
  // Z_sender rows: all 32 lanes cooperate (lane == channel)
#pragma unroll 1
  for (int e2 = 0; e2 < 16; ++e2)
    z_w[e2 * 32 + lane] = w_z2k[(size_t)zidx_w[e2] * 32 + lane];

  __syncthreads();  // weights visible to all waves

  // ---- per-l: 3 GEMM layers (WMMA) + atomic scatter ----
  const int koff = (lane < 16) ? 0 : 1;  // K pair index within a 4-wide step
  const int klo  = koff * 2;
#pragma unroll 1
  for (int l = 0; l < 4; ++l) {
    const float* W0 = s_w0 + l * 256;
    const float* W1 = s_w1 + l * 1024;
    const float* W2 = s_w2 + l * 1024;

    // layer 0: h = silu(rb @ W0 + b0), K=8
    {
      v8f acc0, acc1;
      const float bb0 = s_b0[l * 32 + l15], bb1 = s_b0[l * 32 + 16 + l15];
#pragma unroll
      for (int j = 0; j < 8; ++j) { acc0[j] = bb0; acc1[j] = bb1; }
#pragma unroll
      for (int s = 0; s < 2; ++s) {
        const float* wrow = W0 + (s * 2 + koff) * 64;
        const v2f a  = ld2(rb_w + l15 * 8 + s * 4 + klo);
        const v2f bA = ld2(wrow + l15 * 2);
        const v2f bB = ld2(wrow + 32 + l15 * 2);
        acc0 = wmma_f32_16x16x4(a, bA, acc0);
        acc1 = wmma_f32_16x16x4(a, bB, acc1);
      }
      const int mo = (lane < 16) ? 0 : 8;
#pragma unroll
      for (int j = 0; j < 8; ++j) {
        h_w[(j + mo) * 32 + l15]      = fast_silu(acc0[j]);
        h_w[(j + mo) * 32 + 16 + l15] = fast_silu(acc1[j]);
      }
    }
    // layer 1: h = silu(h @ W1 + b1), K=32 (in-place through LDS)
    mlp_layer_k32(W1, s_b1 + l * 32, h_w, lane, l15, klo, koff, true);
    // layer 2: Rl = h @ W2 + b2, K=32 (no activation)
    mlp_layer_k32(W2, s_b2 + l * 32, h_w, lane, l15, klo, koff, false);

    // scatter: A[recv][l^2 + m][k] += Rl[e][k] * Ylm[e][l^2+m] * Z[e][k]
    const int loff = l * l;
    const int nm = 2 * l + 1;
#pragma unroll 1
    for (int e2 = 0; e2 < 16; ++e2) {
      const int eg = tileBase + e2;
      if (eg < E) {
        const int r = recv_w[e2];
        const float rz = h_w[e2 * 32 + lane] * z_w[e2 * 32 + lane];
        float* dst = A_ws + ((size_t)r * 16 + loff) * 32 + lane;
        for (int m = 0; m < nm; ++m)
          atomicAdd(dst + (size_t)m * 32, rz * ylm_w[e2 * 16 + loff + m]);
      }
    }
  }
}

// ---------------------------------------------------------------------------
// Node kernel: per-l channel mix + order-2 symmetric contraction.
// One wave per node; lane == channel k.
// ---------------------------------------------------------------------------
#define NODE_SMEM_FLOATS (4096 + 256 + 8 * 512 + 8 * 512)
#define NODE_SMEM_BYTES  (NODE_SMEM_FLOATS * 4)

__global__ __launch_bounds__(256) void ace_node_kernel(
    const float* __restrict__ A_ws, const float* __restrict__ cg,
    const int* __restrict__ atomic_num, const float* __restrict__ mix_w,
    const float* __restrict__ sw10, const float* __restrict__ sw11,
    const float* __restrict__ sw20, const float* __restrict__ sw21,
    float* __restrict__ Bout, int Nnodes) {
  extern __shared__ float smem[];
  float* s_mix = smem;                 // 4*32*32
  float* s_cg  = smem + 4096;          // 243 (padded 256)
  float* s_a   = smem + 4096 + 256;    // 8 * 512
  float* s_am  = s_a + 8 * 512;        // 8 * 512

  const int tid = threadIdx.x;
  for (int i = tid; i < 4096; i += 256) s_mix[i] = mix_w[i];
  for (int i = tid; i < 243; i += 256) s_cg[i] = cg[i];
  __syncthreads();

  const int wave = tid >> 5, lane = tid & 31;
  const int n = blockIdx.x * 8 + wave;
  if (n >= Nnodes) return;

  float* at = s_a + wave * 512;
  float* am = s_am + wave * 512;
#pragma unroll
  for (int m = 0; m < 16; ++m)
    at[m * 32 + lane] = A_ws[((size_t)n * 16 + m) * 32 + lane];

  // per-l channel mixing: Amix[m,k] = sum_c A[m,c] * mix_w[l(m)][c,k]
#pragma unroll 1
  for (int m = 0; m < 16; ++m) {
    const int l = (m == 0) ? 0 : (m < 4) ? 1 : (m < 9) ? 2 : 3;
    const float* W = s_mix + l * 1024;
    float s = 0.0f;
#pragma unroll
    for (int c = 0; c < 32; ++c) s += at[m * 32 + c] * W[c * 32 + lane];
    am[m * 32 + lane] = s;
  }

  const int z = atomic_num[n];
  const int LOFF[5] = {0, 1, 4, 9, 16};

  // B0: sc_w1_0[z]*A0 + sum_p sc_w2_0[z,p] * (A_p . A_p via CG0[p])
  float b0v = sw10[(size_t)z * 32 + lane] * am[lane];
  {
    const int CG0_OFF[4] = {0, 1, 10, 35};
#pragma unroll 1
    for (int p = 0; p < 4; ++p) {
      const int d = 2 * p + 1, off = LOFF[p];
      const float* cgp = s_cg + CG0_OFF[p];
      float t = 0.0f;
      for (int i = 0; i < d; ++i) {
        const float ai = am[(off + i) * 32 + lane];
        for (int j = 0; j < d; ++j)
          t += ai * am[(off + j) * 32 + lane] * cgp[i * d + j];
      }
      b0v += sw20[((size_t)z * 4 + p) * 32 + lane] * t;
    }
  }
  Bout[(size_t)n * 128 + lane] = b0v;

  // B1: sc_w1_1[z]*A1 + sum_p sc_w2_1[z,p] * (A_a x A_b via CG1[p])
  float b1v0 = sw11[(size_t)z * 32 + lane] * am[1 * 32 + lane];
  float b1v1 = sw11[(size_t)z * 32 + lane] * am[2 * 32 + lane];
  float b1v2 = sw11[(size_t)z * 32 + lane] * am[3 * 32 + lane];
  {
    const int CG1_OFF[3] = {84, 93, 138};
    const int PA[3] = {0, 1, 2}, PB[3] = {1, 2, 3};
#pragma unroll 1
    for (int p = 0; p < 3; ++p) {
      const int da = 2 * PA[p] + 1, db = 2 * PB[p] + 1;
      const int offa = LOFF[PA[p]], offb = LOFF[PB[p]];
      const float* cgp = s_cg + CG1_OFF[p];
      float t0 = 0.0f, t1 = 0.0f, t2 = 0.0f;
      for (int i = 0; i < da; ++i) {
        const float ai = am[(offa + i) * 32 + lane];
        for (int j = 0; j < db; ++j) {
          const float aij = ai * am[(offb + j) * 32 + lane];
          const float* cc = cgp + (i * db + j) * 3;
          t0 += aij * cc[0]; t1 += aij * cc[1]; t2 += aij * cc[2];
        }
      }
      const float w = sw21[((size_t)z * 3 + p) * 32 + lane];
      b1v0 += w * t0; b1v1 += w * t1; b1v2 += w * t2;
    }
  }
  Bout[(size_t)n * 128 + 32 + lane] = b1v0;
  Bout[(size_t)n * 128 + 64 + lane] = b1v1;
  Bout[(size_t)n * 128 + 96 + lane] = b1v2;
}

// ---------------------------------------------------------------------------
extern "C" void kernel_launch(void* const* d_in, const int* in_sizes, int n_in,
                              void* d_out, int out_size, void* d_ws,
                              size_t ws_size, hipStream_t stream) {
  (void)n_in; (void)out_size; (void)ws_size;
  const float* pos    = (const float*)d_in[0];
  const int*   an     = (const int*)d_in[1];
  const int*   ei     = (const int*)d_in[2];
  const float* shifts = (const float*)d_in[3];
  const float* w_z2k  = (const float*)d_in[4];
  const float* w0     = (const float*)d_in[5];
  const float* b0     = (const float*)d_in[6];
  const float* w1     = (const float*)d_in[7];
  const float* b1     = (const float*)d_in[8];
  const float* w2     = (const float*)d_in[9];
  const float* b2     = (const float*)d_in[10];
  const float* mixw   = (const float*)d_in[11];
  const float* sw10   = (const float*)d_in[12];
  const float* sw11   = (const float*)d_in[13];
  const float* sw20   = (const float*)d_in[14];
  const float* sw21   = (const float*)d_in[15];

  const int N = in_sizes[0] / 3;
  const int E = in_sizes[2] / 2;

  float* A_ws = (float*)d_ws;                     // N*16*32 floats
  float* cg   = A_ws + (size_t)N * 512;           // 243 floats
  float* Bout    = (float*)d_out;                 // N*128
  float* rb_out  = Bout + (size_t)N * 128;        // E*8
  float* ylm_out = rb_out + (size_t)E * 8;        // E*16

  ace_zero_kernel<<<2048, 256, 0, stream>>>(A_ws, (long long)N * 512);
  ace_cg_init_kernel<<<1, 32, 0, stream>>>(cg);

  if (E > 0) {
    const int eblocks = (E + 127) / 128;
    ace_edge_kernel<<<eblocks, 256, EDGE_SMEM_BYTES, stream>>>(
        pos, an, ei, shifts, w_z2k, w0, b0, w1, b1, w2, b2,
        A_ws, rb_out, ylm_out, E);
  }
  if (N > 0) {
    const int nblocks = (N + 7) / 8;
    ace_node_kernel<<<nblocks, 256, NODE_SMEM_BYTES, stream>>>(
        A_ws, cg, an, mixw, sw10, sw11, sw20, sw21, Bout, N);
  }
}